// yolov2_loss_fast_36103495090634
// MI455X (gfx1250) — compile-verified
//
#include <hip/hip_runtime.h>
#include <hip/hip_bf16.h>

// YOLOv2 loss on gfx1250: single-pass streaming reduction (memory-bound,
// ~185 MB @ 23.3 TB/s ~ 8 us). f32 throughout. Cross-lane reduction done with
// V_WMMA_F32_16X16X4_F32 (full-precision f32 WMMA) instead of shuffles.
// Deterministic: per-block partials in d_ws, fixed-order finalize kernel.

typedef float v2f __attribute__((ext_vector_type(2)));
typedef float v8f __attribute__((ext_vector_type(8)));

namespace {

constexpr int BDIM = 256;
constexpr int Bn = 32, An = 5, Kn = 20, Hn = 76, Wn = 76;
constexpr int HW = Hn * Wn;          // 5776
constexpr int NCELL = Bn * HW;       // 184832
constexpr float L_COORD = 5.0f, L_OBJ = 1.0f, L_NOOBJ = 0.5f;

__device__ __forceinline__ float smooth_l1(float d) {
    float a = fabsf(d);
    return (a < 1.0f) ? 0.5f * a * a : a - 0.5f;
}

} // namespace

__global__ void yolo_loss_main(const float* __restrict__ pc,   // pred_cls   [B, A*K, H, W]
                               const float* __restrict__ pr,   // pred_resp  [B, A, H, W]
                               const float* __restrict__ pb,   // pred_bbox  [B, A*4, H, W]
                               const float* __restrict__ lc,   // label_cls  [B, A*K, H, W]
                               const float* __restrict__ lr,   // label_resp [B, A, H, W]
                               const float* __restrict__ lb,   // label_bbox [B, A*4, H, W]
                               float* __restrict__ partials)   // [gridDim.x * 4]
{
    float s0 = 0.f, s1 = 0.f, s2 = 0.f, s3 = 0.f;   // pObj, nObj, cls, ofx (raw sums)

    for (int cell = blockIdx.x * BDIM + threadIdx.x; cell < NCELL;
         cell += gridDim.x * BDIM) {
        const int b  = cell / HW;
        const int hw = cell - b * HW;
        const int baseR = b * An * HW + hw;

        float lrv[An], prv[An];
        float sumlr = 0.f;
        #pragma unroll
        for (int a = 0; a < An; ++a) {
            lrv[a] = lr[baseR + a * HW];
            prv[a] = pr[baseR + a * HW];
            sumlr += lrv[a];
            if (lrv[a] < 1.0f) s1 += smooth_l1(prv[a] - lrv[a]);   // noobj term
        }

        if (sumlr > 0.9f) {   // positive cell (~10%): only now touch bbox/cls
            float best = -1.0f;
            int   bi   = 0;
            float bpo[4], blo[4];
            #pragma unroll
            for (int a = 0; a < An; ++a) {
                const int baseB = (b * An + a) * 4 * HW + hw;
                const float l0 = lb[baseB], l1v = lb[baseB + HW];
                const float l2 = lb[baseB + 2 * HW], l3 = lb[baseB + 3 * HW];
                const float p0 = pb[baseB], p1v = pb[baseB + HW];
                const float p2 = pb[baseB + 2 * HW], p3 = pb[baseB + 3 * HW];
                // corners (mirrors reference _corners)
                const float tx1 = l0 - l2 * 0.5f, ty1 = l1v - l3 * 0.5f;
                const float tx2 = tx1 + l2,       ty2 = ty1 + l3;
                const float px1 = p0 - p2 * 0.5f, py1 = p1v - p3 * 0.5f;
                const float px2 = px1 + p2,       py2 = py1 + p3;
                const float x1 = fmaxf(tx1, px1), y1 = fmaxf(ty1, py1);
                const float x2 = fminf(tx2, px2), y2 = fminf(ty2, py2);
                const float inter = fmaxf(x2 - x1, 0.f) * fmaxf(y2 - y1, 0.f);
                const float at = (tx2 - tx1) * (ty2 - ty1);
                const float ap = (px2 - px1) * (py2 - py1);
                float iou = inter / (at + ap - inter + 1e-10f);
                const float mlo = (l0 + l1v + l2 + l3) * 0.25f;
                if (mlo > 0.f && iou < 0.01f) iou = 0.01f;
                if (iou > best) {               // strict '>' => first max (jnp.argmax)
                    best = iou; bi = a;
                    bpo[0] = p0; bpo[1] = p1v; bpo[2] = p2; bpo[3] = p3;
                    blo[0] = l0; blo[1] = l1v; blo[2] = l2; blo[3] = l3;
                }
            }
            s0 += smooth_l1(prv[bi] - best);
            #pragma unroll
            for (int j = 0; j < 4; ++j) s3 += smooth_l1(bpo[j] - blo[j]);

            const int baseC = (b * An + bi) * Kn * HW + hw;
            #pragma unroll 4
            for (int k = 0; k < Kn; ++k) {
                const float d = pc[baseC + k * HW] - lc[baseC + k * HW];
                s2 += d * d;
            }
        }
    }

    // ---- wave-level reduction via V_WMMA_F32_16X16X4_F32 (full f32) ----
    // A(16x4): lane m (0-15) supplies A[m][0],A[m][1]; lane m+16 supplies A[m][2],A[m][3].
    // Selector B: column n receives component n:
    //   WMMA1 (s0,s1): B[0][0]=B[2][0]=1, B[1][1]=B[3][1]=1
    //   WMMA2 (s2,s3): B[0][2]=B[2][2]=1, B[1][3]=B[3][3]=1  (accumulates into same C)
    // => D[m][n] = component-n sum over lanes {m, m+16}.
    const int lane = threadIdx.x & 31;
    const int l15  = lane & 15;
    v2f a1; a1.x = s0; a1.y = s1;
    v2f a2; a2.x = s2; a2.y = s3;
    v2f b1; b1.x = (l15 == 0) ? 1.f : 0.f; b1.y = (l15 == 1) ? 1.f : 0.f;
    v2f b2; b2.x = (l15 == 2) ? 1.f : 0.f; b2.y = (l15 == 3) ? 1.f : 0.f;
    v8f c = {};
    c = __builtin_amdgcn_wmma_f32_16x16x4_f32(false, a1, false, b1, (short)0, c, false, false);
    c = __builtin_amdgcn_wmma_f32_16x16x4_f32(false, a2, false, b2, (short)0, c, false, false);
    // lane n holds D[m][n] for 8 values of m in its 8 C VGPRs; sum them,
    // then fold lanes n <-> n+16 to cover all 16 rows.
    float t = c[0] + c[1] + c[2] + c[3] + c[4] + c[5] + c[6] + c[7];
    t += __shfl_xor(t, 16, 32);
    // lanes 0..3 now hold the wave's component sums.

    __shared__ float wsum[(BDIM / 32) * 4];
    const int wave = threadIdx.x >> 5;
    if (lane < 4) wsum[wave * 4 + lane] = t;
    __syncthreads();
    if (threadIdx.x < 4) {
        float acc = 0.f;
        #pragma unroll
        for (int w = 0; w < BDIM / 32; ++w) acc += wsum[w * 4 + threadIdx.x];
        partials[blockIdx.x * 4 + threadIdx.x] = acc;
    }
}

__global__ void yolo_loss_finalize(const float* __restrict__ partials,
                                   float* __restrict__ out, int nb)
{
    __shared__ float red[256];
    const int t = threadIdx.x;
    const int comp = t & 3;
    const int chunk = t >> 2;                 // 64 chunks per component
    float s = 0.f;
    for (int i = chunk; i < nb; i += 64) s += partials[i * 4 + comp];
    red[t] = s;
    __syncthreads();
    for (int off = 128; off >= 4; off >>= 1) {   // fixed-order tree (deterministic)
        if (t < off) red[t] += red[t + off];
        __syncthreads();
    }
    if (t < 4) {
        float v = red[t] * (1.0f / (float)Bn);
        if (t == 0) v *= L_OBJ;
        if (t == 1) v *= L_NOOBJ;
        if (t == 3) v *= L_COORD;
        out[t] = v;
    }
}

extern "C" void kernel_launch(void* const* d_in, const int* in_sizes, int n_in,
                              void* d_out, int out_size, void* d_ws, size_t ws_size,
                              hipStream_t stream) {
    (void)in_sizes; (void)n_in; (void)out_size;
    const float* pc = (const float*)d_in[0];
    const float* pr = (const float*)d_in[1];
    const float* pb = (const float*)d_in[2];
    const float* lc = (const float*)d_in[3];
    const float* lr = (const float*)d_in[4];
    const float* lb = (const float*)d_in[5];
    float* out      = (float*)d_out;
    float* partials = (float*)d_ws;

    int nb = (NCELL + BDIM - 1) / BDIM;                       // 722 blocks
    const int ws_cap = (int)(ws_size / (4 * sizeof(float)));
    if (nb > ws_cap) nb = (ws_cap > 0) ? ws_cap : 1;

    yolo_loss_main<<<nb, BDIM, 0, stream>>>(pc, pr, pb, lc, lr, lb, partials);
    yolo_loss_finalize<<<1, 256, 0, stream>>>(partials, out, nb);
}